// SceneGraph3D_15633680957638
// MI455X (gfx1250) — compile-verified
//
#include <hip/hip_runtime.h>

typedef float v2f __attribute__((ext_vector_type(2)));
typedef float v4f __attribute__((ext_vector_type(4)));
typedef float v8f __attribute__((ext_vector_type(8)));

#define XSTRIDE 260   // even (b64 LDS align), mult of 4 floats (b128 align), 4-bank row skew

// -----------------------------------------------------------------------------
// Pack a [256][256] row-major weight W (k-major) into WMMA B-fragment order:
//   dst[((col*2 + hi)*64 + g)*2 + e] = W[4g + 2hi + e][col]
// so lane (hi,l) streams its fragments contiguously; one b128 = 2 K-steps.
// -----------------------------------------------------------------------------
__global__ void pack_kernel(const float* __restrict__ src, float* __restrict__ dst) {
    int id = blockIdx.x * 256 + threadIdx.x;     // 65536 elements
    int e   = id & 1;
    int g   = (id >> 1) & 63;
    int hi  = (id >> 7) & 1;
    int col = id >> 8;
    dst[id] = src[(4 * g + 2 * hi + e) * 256 + col];
}

// -----------------------------------------------------------------------------
// Shared WMMA accumulate: acc[t] += X(16x256, LDS) @ W(256x256, packed frag order)
// One wave, all 16 N-tiles in registers. 2 K-steps per b128 weight load.
// -----------------------------------------------------------------------------
__device__ __forceinline__ void wmma_accumulate(
    const float* X, const float* __restrict__ WP, int hi, int l, v8f acc[16]) {
    #pragma unroll 1
    for (int gg = 0; gg < 64; gg += 2) {
        const int kk = gg << 2;
        v2f a0 = *(const v2f*)(&X[l * XSTRIDE + kk + (hi << 1)]);
        v2f a1 = *(const v2f*)(&X[l * XSTRIDE + kk + 4 + (hi << 1)]);
        v4f w[16];
        #pragma unroll
        for (int t = 0; t < 16; ++t)
            w[t] = *(const v4f*)(WP + (t * 16 + l) * 256 + (hi << 7) + (gg << 1));
        #pragma unroll
        for (int t = 0; t < 16; ++t)
            acc[t] = __builtin_amdgcn_wmma_f32_16x16x4_f32(
                false, a0, false, __builtin_shufflevector(w[t], w[t], 0, 1),
                (short)0, acc[t], false, false);
        #pragma unroll
        for (int t = 0; t < 16; ++t)
            acc[t] = __builtin_amdgcn_wmma_f32_16x16x4_f32(
                false, a1, false, __builtin_shufflevector(w[t], w[t], 2, 3),
                (short)0, acc[t], false, false);
    }
}

// -----------------------------------------------------------------------------
// part kernel: f = relu(pn) [2048,256]; part_i = f @ W1[:256], part_j = f @ W1[256:]
// One wave per 16-row tile. W1Pi/W1Pj are the two packed halves of W1.
// -----------------------------------------------------------------------------
__global__ __launch_bounds__(64) void part_kernel(
    const float* __restrict__ pn, const float* __restrict__ W1Pi,
    const float* __restrict__ W1Pj,
    float* __restrict__ part_i, float* __restrict__ part_j) {
    __shared__ float Xs[2][16 * XSTRIDE];
    const int wave = threadIdx.x >> 5;
    const int lane = threadIdx.x & 31;
    const int hi   = lane >> 4;
    const int l    = lane & 15;
    float* X = Xs[wave];
    const int tile = blockIdx.x * 2 + wave;   // 128 tiles of 16 rows
    const int row0 = tile * 16;

    // Stage relu(pn) rows into LDS
    for (int idx = lane; idx < 16 * 64; idx += 32) {
        int b  = idx >> 6;
        int c4 = (idx & 63) << 2;
        v4f v = *(const v4f*)(pn + (row0 + b) * 256 + c4);
        v.x = fmaxf(v.x, 0.f); v.y = fmaxf(v.y, 0.f);
        v.z = fmaxf(v.z, 0.f); v.w = fmaxf(v.w, 0.f);
        *(v4f*)(&X[b * XSTRIDE + c4]) = v;
    }
    __syncthreads();

    #pragma unroll 1
    for (int p = 0; p < 2; ++p) {
        float* outp = p ? part_j : part_i;
        const float* WP = p ? W1Pj : W1Pi;
        v8f acc[16];
        #pragma unroll
        for (int t = 0; t < 16; ++t)
            #pragma unroll
            for (int r = 0; r < 8; ++r) acc[t][r] = 0.0f;

        wmma_accumulate(X, WP, hi, l, acc);

        #pragma unroll
        for (int t = 0; t < 16; ++t) {
            int n = t * 16 + l;
            #pragma unroll
            for (int r = 0; r < 8; ++r) {
                int b = r + (hi << 3);
                outp[(row0 + b) * 256 + n] = acc[t][r];
            }
        }
    }
}

// -----------------------------------------------------------------------------
// Fused 3-layer relation chain. One wave per (i,j) pair; B=16 batch rows = one
// WMMA M-tile. All 16 N-tile accumulators live in registers across the K sweep,
// so the LDS activation tile can be overwritten in place between layers.
// -----------------------------------------------------------------------------
__device__ __forceinline__ void wmma_layer256(
    float* X, const float* __restrict__ WP, const float* __restrict__ bias,
    float* __restrict__ gout, bool relu_to_lds, int hi, int l) {
    v8f acc[16];
    #pragma unroll
    for (int t = 0; t < 16; ++t)
        #pragma unroll
        for (int r = 0; r < 8; ++r) acc[t][r] = 0.0f;

    wmma_accumulate(X, WP, hi, l, acc);

    #pragma unroll
    for (int t = 0; t < 16; ++t) {
        int n = t * 16 + l;
        float bv = bias[n];
        #pragma unroll
        for (int r = 0; r < 8; ++r) {
            int b = r + (hi << 3);
            float v = acc[t][r] + bv;
            if (gout) gout[b * 256 + n] = v;
            if (relu_to_lds) X[b * XSTRIDE + n] = fmaxf(v, 0.f);
        }
    }
}

__global__ __launch_bounds__(64) void relation_kernel(
    const float* __restrict__ part_i, const float* __restrict__ part_j,
    const float* __restrict__ b1,
    const float* __restrict__ W2P, const float* __restrict__ b2,
    const float* __restrict__ W3P, const float* __restrict__ b3,
    const float* __restrict__ W4P, const float* __restrict__ b4,
    float* __restrict__ rel_out, float* __restrict__ multi_out) {
    __shared__ float Xs[2][16 * XSTRIDE];
    const int wave = threadIdx.x >> 5;
    const int lane = threadIdx.x & 31;
    const int hi   = lane >> 4;
    const int l    = lane & 15;
    float* X = Xs[wave];
    const int pair = blockIdx.x * 2 + wave;   // 16384 pairs
    const int i = pair >> 7;
    const int j = pair & 127;

    // X1 = relu(part_i[i] + part_j[j] + b1)   -> 16 x 256 tile in LDS
    const float* pi = part_i + (i * 16) * 256;
    const float* pj = part_j + (j * 16) * 256;
    for (int idx = lane; idx < 16 * 64; idx += 32) {
        int b  = idx >> 6;
        int c4 = (idx & 63) << 2;
        v4f a  = *(const v4f*)(pi + b * 256 + c4);
        v4f bb = *(const v4f*)(pj + b * 256 + c4);
        v4f bi = *(const v4f*)(b1 + c4);
        v4f s = a + bb + bi;
        s.x = fmaxf(s.x, 0.f); s.y = fmaxf(s.y, 0.f);
        s.z = fmaxf(s.z, 0.f); s.w = fmaxf(s.w, 0.f);
        *(v4f*)(&X[b * XSTRIDE + c4]) = s;
    }
    __syncthreads();

    // rel = X1 @ W2 + b2  (written pre-relu; relu(rel) stays in LDS)
    wmma_layer256(X, W2P, b2, rel_out + pair * 4096, true, hi, l);
    __syncthreads();
    // m1 = relu(rel) @ W3 + b3  (only relu(m1) kept in LDS)
    wmma_layer256(X, W3P, b3, nullptr, true, hi, l);
    __syncthreads();
    // multi = relu(m1) @ W4 + b4
    wmma_layer256(X, W4P, b4, multi_out + pair * 4096, false, hi, l);
}

// -----------------------------------------------------------------------------
// Object MLP: 256 -> 128(relu) -> 256(relu) -> 512. One block per row.
// -----------------------------------------------------------------------------
__global__ __launch_bounds__(256) void obj_kernel(
    const float* __restrict__ xin,
    const float* __restrict__ Wo1, const float* __restrict__ bo1,
    const float* __restrict__ Wo2, const float* __restrict__ bo2,
    const float* __restrict__ Wo3, const float* __restrict__ bo3,
    float* __restrict__ out) {
    __shared__ float x[256];
    __shared__ float h1[128];
    __shared__ float h2[256];
    const int row = blockIdx.x;
    const int t = threadIdx.x;

    x[t] = xin[row * 256 + t];
    __syncthreads();

    if (t < 128) {
        float s = bo1[t];
        for (int k = 0; k < 256; ++k) s += x[k] * Wo1[k * 128 + t];
        h1[t] = fmaxf(s, 0.f);
    }
    __syncthreads();

    {
        float s = bo2[t];
        for (int k = 0; k < 128; ++k) s += h1[k] * Wo2[k * 256 + t];
        h2[t] = fmaxf(s, 0.f);
    }
    __syncthreads();

    #pragma unroll
    for (int cc = 0; cc < 2; ++cc) {
        int c = t + cc * 256;
        float s = bo3[c];
        for (int k = 0; k < 256; ++k) s += h2[k] * Wo3[k * 512 + c];
        out[row * 512 + c] = s;
    }
}

// -----------------------------------------------------------------------------
extern "C" void kernel_launch(void* const* d_in, const int* in_sizes, int n_in,
                              void* d_out, int out_size, void* d_ws, size_t ws_size,
                              hipStream_t stream) {
    const float* pn  = (const float*)d_in[0];
    const float* enc = (const float*)d_in[1];
    const float* W1  = (const float*)d_in[2];
    const float* b1  = (const float*)d_in[3];
    const float* W2  = (const float*)d_in[4];
    const float* b2  = (const float*)d_in[5];
    const float* W3  = (const float*)d_in[6];
    const float* b3  = (const float*)d_in[7];
    const float* W4  = (const float*)d_in[8];
    const float* b4  = (const float*)d_in[9];
    const float* Wo1 = (const float*)d_in[10];
    const float* bo1 = (const float*)d_in[11];
    const float* Wo2 = (const float*)d_in[12];
    const float* bo2 = (const float*)d_in[13];
    const float* Wo3 = (const float*)d_in[14];
    const float* bo3 = (const float*)d_in[15];

    float* out   = (float*)d_out;
    float* obj   = out;                 // [128,16,512]   = 1,048,576
    float* rel   = out + 1048576;       // [128,128,16,256]
    float* multi = rel + 67108864;      // [128,128,16,256]

    float* ws     = (float*)d_ws;
    float* W2P    = ws;                 // 5 packed 256x256 weight blocks
    float* W3P    = W2P + 65536;
    float* W4P    = W3P + 65536;
    float* W1Pi   = W4P + 65536;
    float* W1Pj   = W1Pi + 65536;
    float* part_i = W1Pj + 65536;       // 2048*256
    float* part_j = part_i + 524288;    // 2048*256

    pack_kernel<<<256, 256, 0, stream>>>(W2, W2P);
    pack_kernel<<<256, 256, 0, stream>>>(W3, W3P);
    pack_kernel<<<256, 256, 0, stream>>>(W4, W4P);
    pack_kernel<<<256, 256, 0, stream>>>(W1, W1Pi);           // rows 0..255
    pack_kernel<<<256, 256, 0, stream>>>(W1 + 65536, W1Pj);   // rows 256..511

    part_kernel<<<64, 64, 0, stream>>>(pn, W1Pi, W1Pj, part_i, part_j);

    relation_kernel<<<8192, 64, 0, stream>>>(part_i, part_j, b1,
                                             W2P, b2, W3P, b3, W4P, b4,
                                             rel, multi);

    obj_kernel<<<2048, 256, 0, stream>>>(enc, Wo1, bo1, Wo2, bo2, Wo3, bo3, obj);
}